// BatchGATLayer_73667279061277
// MI455X (gfx1250) — compile-verified
//
#include <hip/hip_runtime.h>
#include <math.h>

// Problem constants (match reference)
#define NNODES 20000
#define TSTEPS 4
#define IN_DIM 128
#define HEADS  4
#define CH     64            // D_MODEL
#define HC     (HEADS * CH)  // 256 projected channels
#define NT     (NNODES * TSTEPS)
#define NEG_SLOPE 0.2f

typedef float v2f __attribute__((ext_vector_type(2)));
typedef float v8f __attribute__((ext_vector_type(8)));

// Native fire-and-forget f32 atomic add (GLOBAL_ATOMIC_ADD_F32, no CAS loop).
__device__ __forceinline__ void atomicAddF32(float* p, float v) {
  unsafeAtomicAdd(p, v);
}

// ---------------------------------------------------------------------------
// Kernel 0: init max buffer / denominators / output(bias)
// ---------------------------------------------------------------------------
__global__ void gat_init(float* __restrict__ out, float* __restrict__ maxb,
                         float* __restrict__ denom, const float* __restrict__ bias) {
  int idx = blockIdx.x * blockDim.x + threadIdx.x;
  if (idx < NT * HEADS) {
    maxb[idx]  = -3.0e38f;
    denom[idx] = 0.0f;
  }
  if (idx < NT * CH) {
    out[idx] = bias[idx & (CH - 1)];
  }
}

// ---------------------------------------------------------------------------
// Kernel 1: h = x @ W via V_WMMA_F32_16X16X4_F32 (fp32-exact path)
// One wave computes a 16x64 output strip (4 accumulators), K=128 in steps of 4.
// A 16x4 f32: lane(0-15)=M, VGPR0 holds K=0|2 (lane halves), VGPR1 K=1|3.
// B 4x16 f32: lane(0-15)=N, symmetric K striping.
// C/D 16x16 f32: 8 VGPRs, lanes 16-31 hold rows M+8.
// ---------------------------------------------------------------------------
__global__ __launch_bounds__(256) void gat_gemm_wmma(const float* __restrict__ x,
                                                     const float* __restrict__ W,
                                                     float* __restrict__ h) {
  const int lane = threadIdx.x & 31;
  const int wave = blockIdx.x * (blockDim.x >> 5) + (threadIdx.x >> 5);
  const int mtile = wave >> 2;        // 16-row tile index
  if (mtile * 16 >= NT) return;       // wave-uniform (EXEC stays all-ones)
  const int m0 = mtile * 16;
  const int n0 = (wave & 3) * 64;     // 0,64,128,192
  const int rowA = m0 + (lane & 15);
  const int koff = (lane >> 4) << 1;  // 0 or 2
  const int colB = n0 + (lane & 15);

  v8f acc0 = {}, acc1 = {}, acc2 = {}, acc3 = {};
  for (int k = 0; k < IN_DIM; k += 4) {
    const int kk = k + koff;
    v2f a;
    a.x = x[rowA * IN_DIM + kk];
    a.y = x[rowA * IN_DIM + kk + 1];
    v2f b0, b1, b2, b3;
    b0.x = W[kk * HC + colB +  0]; b0.y = W[(kk + 1) * HC + colB +  0];
    b1.x = W[kk * HC + colB + 16]; b1.y = W[(kk + 1) * HC + colB + 16];
    b2.x = W[kk * HC + colB + 32]; b2.y = W[(kk + 1) * HC + colB + 32];
    b3.x = W[kk * HC + colB + 48]; b3.y = W[(kk + 1) * HC + colB + 48];
    acc0 = __builtin_amdgcn_wmma_f32_16x16x4_f32(false, a, false, b0, (short)0, acc0, false, false);
    acc1 = __builtin_amdgcn_wmma_f32_16x16x4_f32(false, a, false, b1, (short)0, acc1, false, false);
    acc2 = __builtin_amdgcn_wmma_f32_16x16x4_f32(false, a, false, b2, (short)0, acc2, false, false);
    acc3 = __builtin_amdgcn_wmma_f32_16x16x4_f32(false, a, false, b3, (short)0, acc3, false, false);
  }

  const int rowHi = (lane >> 4) << 3;  // 0 or 8
  const int col = n0 + (lane & 15);
#pragma unroll
  for (int j = 0; j < 8; ++j) {
    const int row = m0 + rowHi + j;
    h[row * HC + col +  0] = acc0[j];
    h[row * HC + col + 16] = acc1[j];
    h[row * HC + col + 32] = acc2[j];
    h[row * HC + col + 48] = acc3[j];
  }
}

// ---------------------------------------------------------------------------
// Kernel 2: per-(row, head) attention dots a_src, a_dst
// ---------------------------------------------------------------------------
__global__ void gat_attn(const float* __restrict__ h,
                         const float* __restrict__ att_src,
                         const float* __restrict__ att_dst,
                         float* __restrict__ asrc, float* __restrict__ adst) {
  int tid = blockIdx.x * blockDim.x + threadIdx.x;
  if (tid >= NT * HEADS) return;
  const int r  = tid / HEADS;
  const int hd = tid - r * HEADS;
  const float* __restrict__ hp = h + r * HC + hd * CH;
  const float* __restrict__ as = att_src + hd * CH;
  const float* __restrict__ ad = att_dst + hd * CH;
  float s1 = 0.0f, s2 = 0.0f;
#pragma unroll 8
  for (int c = 0; c < CH; ++c) {
    const float v = hp[c];
    s1 += v * as[c];
    s2 += v * ad[c];
  }
  asrc[tid] = s1;
  adst[tid] = s2;
}

// Sign-split float atomic max using native integer atomics (correct for all
// sign combinations; buffer initialized to -3e38).
__device__ __forceinline__ void atomicMaxFloat(float* addr, float val) {
  if (val >= 0.0f) atomicMax((int*)addr, __float_as_int(val));
  else             atomicMin((unsigned int*)addr, __float_as_uint(val));
}

__device__ __forceinline__ float leaky(float l) {
  return l > 0.0f ? l : NEG_SLOPE * l;
}

// ---------------------------------------------------------------------------
// Kernel 3: segment max of edge logits (one thread per (edge, t))
// ---------------------------------------------------------------------------
__global__ void gat_edge_max(const int* __restrict__ src, const int* __restrict__ dst, int E,
                             const float* __restrict__ asrc, const float* __restrict__ adst,
                             float* __restrict__ maxb) {
  int tid = blockIdx.x * blockDim.x + threadIdx.x;
  if (tid >= E * TSTEPS) return;
  const int e = tid >> 2;            // TSTEPS == 4
  const int t = tid & 3;
  const int sb = (src[e] * TSTEPS + t) * HEADS;
  const int db = (dst[e] * TSTEPS + t) * HEADS;
#pragma unroll
  for (int hd = 0; hd < HEADS; ++hd) {
    atomicMaxFloat(&maxb[db + hd], leaky(asrc[sb + hd] + adst[db + hd]));
  }
}

// ---------------------------------------------------------------------------
// Kernel 4: softmax denominators (one thread per (edge, t))
// ---------------------------------------------------------------------------
__global__ void gat_edge_expsum(const int* __restrict__ src, const int* __restrict__ dst, int E,
                                const float* __restrict__ asrc, const float* __restrict__ adst,
                                const float* __restrict__ maxb, float* __restrict__ denom) {
  int tid = blockIdx.x * blockDim.x + threadIdx.x;
  if (tid >= E * TSTEPS) return;
  const int e = tid >> 2;
  const int t = tid & 3;
  const int sb = (src[e] * TSTEPS + t) * HEADS;
  const int db = (dst[e] * TSTEPS + t) * HEADS;
#pragma unroll
  for (int hd = 0; hd < HEADS; ++hd) {
    const float l = leaky(asrc[sb + hd] + adst[db + hd]);
    atomicAddF32(&denom[db + hd], __expf(l - maxb[db + hd]));
  }
}

// ---------------------------------------------------------------------------
// Kernel 5: message aggregation. One wave per (edge, t); head-mean folded into
// alpha; lanes cover the 64 output channels (2 each, coalesced atomics).
// ---------------------------------------------------------------------------
__global__ __launch_bounds__(256) void gat_edge_aggregate(
    const int* __restrict__ src, const int* __restrict__ dst, int E,
    const float* __restrict__ asrc, const float* __restrict__ adst,
    const float* __restrict__ maxb, const float* __restrict__ denom,
    const float* __restrict__ h, float* __restrict__ out) {
  const int lane = threadIdx.x & 31;
  const int w = blockIdx.x * (blockDim.x >> 5) + (threadIdx.x >> 5);
  if (w >= E * TSTEPS) return;       // wave-uniform
  const int e = w >> 2;
  const int t = w & 3;
  const int s = src[e];
  const int d = dst[e];
  const int sb = (s * TSTEPS + t) * HEADS;
  const int db = (d * TSTEPS + t) * HEADS;

  float alpha[HEADS];
#pragma unroll
  for (int hd = 0; hd < HEADS; ++hd) {
    const float l  = leaky(asrc[sb + hd] + adst[db + hd]);
    const float ex = __expf(l - maxb[db + hd]);
    alpha[hd] = ex / (denom[db + hd] + 1e-16f) * (1.0f / HEADS);
  }

  const float* __restrict__ hp = h + (s * TSTEPS + t) * HC;
  float* __restrict__ op = out + (d * TSTEPS + t) * CH;
#pragma unroll
  for (int half = 0; half < 2; ++half) {
    const int c = lane + 32 * half;
    const float m = alpha[0] * hp[c]
                  + alpha[1] * hp[CH + c]
                  + alpha[2] * hp[2 * CH + c]
                  + alpha[3] * hp[3 * CH + c];
    atomicAddF32(&op[c], m);
  }
}

// ---------------------------------------------------------------------------
// Launch wrapper
// ---------------------------------------------------------------------------
extern "C" void kernel_launch(void* const* d_in, const int* in_sizes, int n_in,
                              void* d_out, int out_size, void* d_ws, size_t ws_size,
                              hipStream_t stream) {
  const float* x       = (const float*)d_in[0];
  const int*   ei      = (const int*)  d_in[1];
  const float* W       = (const float*)d_in[2];
  const float* att_src = (const float*)d_in[3];
  const float* att_dst = (const float*)d_in[4];
  const float* bias    = (const float*)d_in[5];
  float* out = (float*)d_out;

  const int E = in_sizes[1] / 2;
  const int* src = ei;
  const int* dst = ei + E;

  // Workspace layout (~87 MB): h | a_src | a_dst | max | denom
  float* hbuf  = (float*)d_ws;
  float* asrc  = hbuf + (size_t)NT * HC;
  float* adst  = asrc + (size_t)NT * HEADS;
  float* maxb  = adst + (size_t)NT * HEADS;
  float* denom = maxb + (size_t)NT * HEADS;

  {
    const int total = NT * CH;  // covers the smaller NT*HEADS ranges too
    gat_init<<<(total + 255) / 256, 256, 0, stream>>>(out, maxb, denom, bias);
  }
  {
    const int waves  = (NT / 16) * 4;        // 20000 waves
    const int blocks = (waves + 7) / 8;      // 8 waves / 256-thread block
    gat_gemm_wmma<<<blocks, 256, 0, stream>>>(x, W, hbuf);
  }
  {
    const int total = NT * HEADS;
    gat_attn<<<(total + 255) / 256, 256, 0, stream>>>(hbuf, att_src, att_dst, asrc, adst);
  }
  {
    const int total = E * TSTEPS;
    gat_edge_max<<<(total + 255) / 256, 256, 0, stream>>>(src, dst, E, asrc, adst, maxb);
    gat_edge_expsum<<<(total + 255) / 256, 256, 0, stream>>>(src, dst, E, asrc, adst, maxb, denom);
  }
  {
    const int waves  = E * TSTEPS;
    const int blocks = (waves + 7) / 8;
    gat_edge_aggregate<<<blocks, 256, 0, stream>>>(src, dst, E, asrc, adst, maxb, denom, hbuf, out);
  }
}